// SPH_CAL_66211215835310
// MI455X (gfx1250) — compile-verified
//
#include <hip/hip_runtime.h>

// ---------------------------------------------------------------------------
// Real spherical harmonics up to l=4 (25 rows) for N points, cart (3,N) f32.
// Memory-bound streaming kernel: 12 B in, 100 B out per point.
// HBM floor on MI455X: ~224 MB / 23.3 TB/s ~= 9.6 us; compute is ~70 fp32
// FLOPs/point -> far below the memory roof, so no matrix-op formulation can
// win. CDNA5 paths used: async global->LDS b128 loads (ASYNCcnt-tracked),
// s_wait_asynccnt, 128-bit non-temporal stores, wave32 256-thread blocks.
// ---------------------------------------------------------------------------

typedef float v4f __attribute__((ext_vector_type(4)));
typedef int v4i __attribute__((ext_vector_type(4)));
typedef __attribute__((address_space(1))) v4i* glb_v4i_ptr; // global (AS1)
typedef __attribute__((address_space(3))) v4i* lds_v4i_ptr; // LDS (AS3)

#ifndef __has_builtin
#define __has_builtin(x) 0
#endif

#if __has_builtin(__builtin_amdgcn_global_load_async_to_lds_b128)
#define SPH_HAVE_ASYNC 1
#else
#define SPH_HAVE_ASYNC 0
#endif

__device__ __forceinline__ void sph_wait_async() {
#if __has_builtin(__builtin_amdgcn_s_wait_asynccnt)
  __builtin_amdgcn_s_wait_asynccnt(0);
#else
  asm volatile("s_wait_asynccnt 0" ::: "memory");
#endif
}

template <typename T> __device__ __forceinline__ T sph_splat(float c);
template <> __device__ __forceinline__ float sph_splat<float>(float c) { return c; }
template <> __device__ __forceinline__ v4f sph_splat<v4f>(float c) {
  v4f r = {c, c, c, c};
  return r;
}

// Constants (match reference sign conventions exactly):
//   row1 = -sqrt(3/(4pi)) * y, row2 = +sqrt(3/(4pi)) * z, row3 = -sqrt(3/(4pi)) * x
// l=3,4 from the recurrence with
//   ca(l,m)=sqrt((4l^2-1)/(l^2-m^2)), cb(l,m)=-sqrt(((l-1)^2-m^2)/(4(l-1)^2-1))
template <typename T>
__device__ __forceinline__ void sph_eval(const T x, const T y, const T z, T s[25]) {
  const T d = x * x + y * y + z * z;
  s[0] = sph_splat<T>(0.28209479177387814f);          // 0.5/sqrt(pi)
  s[1] = -0.4886025119029199f * y;
  s[2] = 0.4886025119029199f * z;
  s[3] = -0.4886025119029199f * x;
  s[4] = 1.0925484305920792f * (x * y);               // sqrt(15/4pi)
  s[5] = -1.0925484305920792f * (y * z);
  s[6] = 0.31539156525252005f * (3.0f * (z * z) - d); // sqrt(5/16pi)
  s[7] = -1.0925484305920792f * (x * z);
  s[8] = 0.5462742152960396f * (x * x - y * y);       // sqrt(15/16pi)
  // ---- l = 3 (p1=-sqrt(7/6), p2[2]=sqrt(7)) ----
  s[9]  = -1.0801234497346435f * (x * s[4] + y * s[8]);
  s[10] = 2.6457513110645907f * (z * s[4]);
  s[11] = 2.0916500663351889f * (z * s[5] - 0.4472135954999579f * (d * s[1]));
  s[12] = 1.9720265943665387f * (z * s[6] - 0.5163977794943222f * (d * s[2]));
  s[13] = 2.0916500663351889f * (z * s[7] - 0.4472135954999579f * (d * s[3]));
  s[14] = 2.6457513110645907f * (z * s[8]);
  s[15] = -1.0801234497346435f * (x * s[8] - y * s[4]);
  // ---- l = 4 (p1=-sqrt(9/8), p2[3]=3) ----
  s[16] = -1.0606601717798212f * (x * s[9] + y * s[15]);
  s[17] = 3.0f * (z * s[9]);
  s[18] = 2.2912878474779199f * (z * s[10] - 0.3779644730092272f * (d * s[4]));
  s[19] = 2.0493901531919194f * (z * s[11] - 0.4780914437337574f * (d * s[5]));
  s[20] = 1.9843134832984430f * (z * s[12] - 0.5070925528371099f * (d * s[6]));
  s[21] = 2.0493901531919194f * (z * s[13] - 0.4780914437337574f * (d * s[7]));
  s[22] = 2.2912878474779199f * (z * s[14] - 0.3779644730092272f * (d * s[8]));
  s[23] = 3.0f * (z * s[15]);
  s[24] = -1.0606601717798212f * (x * s[15] - y * s[9]);
}

#define SPH_BLK 256
#define SPH_VEC 4
#define SPH_TILE (SPH_BLK * SPH_VEC) // 1024 points per block

__global__ __launch_bounds__(SPH_BLK) void sph_cal_kernel(
    const float* __restrict__ cart, float* __restrict__ out, long long N,
    int fullBlocks) {
#if SPH_HAVE_ASYNC
  __shared__ __align__(16) float tx[SPH_TILE];
  __shared__ __align__(16) float ty[SPH_TILE];
  __shared__ __align__(16) float tz[SPH_TILE];
#endif
  const long long base = (long long)blockIdx.x * SPH_TILE;

  if ((int)blockIdx.x < fullBlocks) {
    // ---- full tile: 4 points per thread, 128-bit data path ----
    const int t4 = threadIdx.x * SPH_VEC;
    const float* gx = cart + base;
    const float* gy = cart + N + base;
    const float* gz = cart + 2 * N + base;
    v4f x, y, z;
#if SPH_HAVE_ASYNC
    // Stage inputs through LDS on the gfx1250 async path (ASYNCcnt-tracked).
    // Each lane owns a private 16B slot; only the wave's own counter gates
    // the readback, so no workgroup barrier is required.
    __builtin_amdgcn_global_load_async_to_lds_b128(
        (glb_v4i_ptr)(gx + t4), (lds_v4i_ptr)&tx[t4], 0, 0);
    __builtin_amdgcn_global_load_async_to_lds_b128(
        (glb_v4i_ptr)(gy + t4), (lds_v4i_ptr)&ty[t4], 0, 0);
    __builtin_amdgcn_global_load_async_to_lds_b128(
        (glb_v4i_ptr)(gz + t4), (lds_v4i_ptr)&tz[t4], 0, 0);
    sph_wait_async();
    asm volatile("" ::: "memory");
    x = *(const v4f*)&tx[t4];
    y = *(const v4f*)&ty[t4];
    z = *(const v4f*)&tz[t4];
#else
    x = *(const v4f*)(gx + t4);
    y = *(const v4f*)(gy + t4);
    z = *(const v4f*)(gz + t4);
#endif
    v4f s[25];
    sph_eval<v4f>(x, y, z, s);
    float* o = out + base + t4;
#pragma unroll
    for (int r = 0; r < 25; ++r) {
      // Output stream (200 MB) is write-once: bypass with non-temporal hint.
      __builtin_nontemporal_store(s[r], (v4f*)(o + (long long)r * N));
    }
  } else {
    // ---- tail (or unaligned-N fallback): scalar per point ----
    long long end = base + SPH_TILE;
    if (end > N) end = N;
    for (long long i = base + threadIdx.x; i < end; i += SPH_BLK) {
      const float x = cart[i];
      const float y = cart[N + i];
      const float z = cart[2 * N + i];
      float s[25];
      sph_eval<float>(x, y, z, s);
#pragma unroll
      for (int r = 0; r < 25; ++r) out[(long long)r * N + i] = s[r];
    }
  }
}

extern "C" void kernel_launch(void* const* d_in, const int* in_sizes, int n_in,
                              void* d_out, int out_size, void* d_ws,
                              size_t ws_size, hipStream_t stream) {
  (void)n_in;
  (void)out_size;
  (void)d_ws;
  (void)ws_size;
  const float* cart = (const float*)d_in[0];
  float* out = (float*)d_out;
  const long long N = (long long)(in_sizes[0] / 3);
  if (N <= 0) return;
  const int grid = (int)((N + SPH_TILE - 1) / SPH_TILE);
  // Vector path requires 16B alignment of each row slice => N % 4 == 0.
  const int fullBlocks = ((N & 3LL) == 0) ? (int)(N / SPH_TILE) : 0;
  sph_cal_kernel<<<grid, SPH_BLK, 0, stream>>>(cart, out, N, fullBlocks);
}